// PPOExternalMemInteractiveModel_52716428591858
// MI455X (gfx1250) — compile-verified
//
#include <hip/hip_runtime.h>
#include <math.h>

typedef __attribute__((ext_vector_type(2))) float v2f;
typedef __attribute__((ext_vector_type(4))) float v4f;
typedef __attribute__((ext_vector_type(8))) float v8f;

#define NB    64      // persistent blocks (co-resident)
#define NT    512     // threads per block = 16 wave32
#define HDIM  256
#define PITCH 36      // LDS stage pitch: 16B-aligned rows, conflict-free column reads
#define NEG_INF (-3.0e38f)

// workspace layout (float offsets; words 0..1 are barrier atomics)
#define WS_XIN  16
#define WS_CELL (WS_XIN + 256)
#define WS_PROC (WS_CELL + 256)
#define WS_MG   (WS_PROC + 256)
#define WS_BMAX (WS_MG + 256)
#define WS_BSUM (WS_BMAX + NB)
#define WS_TKV  (WS_BSUM + NB)
#define WS_TKI  (WS_TKV + NB*10)

__device__ __forceinline__ float sigmf(float x) { return 1.0f / (1.0f + __expf(-x)); }

// sense-reversing grid barrier (all NB blocks co-resident by construction)
__device__ __forceinline__ void grid_sync(unsigned* cnt, unsigned* gen)
{
  __syncthreads();
  if (threadIdx.x == 0) {
    __threadfence();
    unsigned g = __hip_atomic_load(gen, __ATOMIC_ACQUIRE, __HIP_MEMORY_SCOPE_AGENT);
    unsigned a = __hip_atomic_fetch_add(cnt, 1u, __ATOMIC_ACQ_REL, __HIP_MEMORY_SCOPE_AGENT);
    if (a == (unsigned)(NB - 1)) {
      __hip_atomic_store(cnt, 0u, __ATOMIC_RELAXED, __HIP_MEMORY_SCOPE_AGENT);
      __hip_atomic_fetch_add(gen, 1u, __ATOMIC_ACQ_REL, __HIP_MEMORY_SCOPE_AGENT);
    } else {
      while (__hip_atomic_load(gen, __ATOMIC_ACQUIRE, __HIP_MEMORY_SCOPE_AGENT) == g)
        __builtin_amdgcn_s_sleep(1);
    }
    __threadfence();
  }
  __syncthreads();
}

// k rounds of block-wide argmax over mv/mi[0..nslots) with winner invalidation.
// Ties prefer the smaller element index (jnp.argmax parity).
__device__ void argmax_rounds(float* mv, int* mi, float* rv, int* ri, int* rs,
                              int nslots, int k, float* outv, int* outi)
{
  const int tid = threadIdx.x;
  for (int r = 0; r < k; ++r) {
    float bv = NEG_INF; int bi = 0x7fffffff; int bs = 0;
    for (int s = tid; s < nslots; s += NT) {
      float v = mv[s]; int ii = mi[s];
      if (v > bv || (v == bv && ii < bi)) { bv = v; bi = ii; bs = s; }
    }
    rv[tid] = bv; ri[tid] = bi; rs[tid] = bs;
    __syncthreads();
    for (int s = NT/2; s > 0; s >>= 1) {
      if (tid < s) {
        if (rv[tid+s] > rv[tid] || (rv[tid+s] == rv[tid] && ri[tid+s] < ri[tid])) {
          rv[tid] = rv[tid+s]; ri[tid] = ri[tid+s]; rs[tid] = rs[tid+s];
        }
      }
      __syncthreads();
    }
    if (tid == 0) { outv[r] = rv[0]; outi[r] = ri[0]; mv[rs[0]] = NEG_INF; }
    __syncthreads();
  }
}

__global__ void ppo_init(const float* __restrict__ um, const float* __restrict__ emb,
                         const int* __restrict__ seq, float* __restrict__ wsF,
                         unsigned* __restrict__ wsU, float* __restrict__ out, int V)
{
  int i = blockIdx.x * blockDim.x + threadIdx.x;
  int T = seq[0];
  float* hist = out + (size_t)T * V + 3 * (size_t)T;   // final_hist slot doubles as working hist
  if (i < V) hist[i] = um[i];
  if (i < HDIM) wsF[WS_XIN + i] = emb[i];              // emb_table[token 0] * s_hit(1.0)
  if (i == 0) { wsU[0] = 0u; wsU[1] = 0u; }
}

__global__ void ppo_procmem(const float* __restrict__ mem, const float* __restrict__ memW,
                            float* __restrict__ wsF, int V)
{
  __shared__ float red[256];
  const int h = blockIdx.x, tid = threadIdx.x;
  const float* row = memW + (size_t)h * V;
  float s = 0.f;
  for (int v = tid; v < V; v += 256) s = fmaf(row[v], mem[v], s);
  red[tid] = s; __syncthreads();
  for (int st = 128; st > 0; st >>= 1) { if (tid < st) red[tid] += red[tid + st]; __syncthreads(); }
  if (tid == 0) wsF[WS_PROC + h] = tanhf(red[0]);
}

__global__ void ppo_mg(const float* __restrict__ fgmW, const float* __restrict__ fgmb,
                       float* __restrict__ wsF)
{
  const int j = threadIdx.x;
  const float* row = fgmW + (size_t)j * HDIM;
  float s = fgmb[j];
  for (int h = 0; h < HDIM; ++h) s = fmaf(row[h], wsF[WS_PROC + h], s);
  wsF[WS_MG + j] = s;   // step-invariant: proc_mem @ fg_mem_W.T + fg_mem_b
}

__global__ __launch_bounds__(NT, 1)
void ppo_main(const float* __restrict__ ui, const float* __restrict__ emb,
              const float* __restrict__ Wih, const float* __restrict__ bih,
              const float* __restrict__ bhh, const float* __restrict__ fcW,
              const float* __restrict__ fcb, const float* __restrict__ valW,
              const float* __restrict__ valb, const float* __restrict__ fgcW,
              const int* __restrict__ seq, float* __restrict__ wsF,
              unsigned* __restrict__ wsU, float* __restrict__ out, int V)
{
  __shared__ alignas(16) float stage[16 * 16 * PITCH];  // per-wave 16x32 fp32 fc_W tiles
  __shared__ float mv[2048];
  __shared__ int   mi[2048];
  __shared__ float rv[NT];
  __shared__ int   ri[NT];
  __shared__ int   rs[NT];
  __shared__ float cls[HDIM];
  __shared__ float wvs[10];
  __shared__ int   wis[10];
  __shared__ float sF[2];
  __shared__ int   sI[2];

  const int tid = threadIdx.x, blk = blockIdx.x;
  const int wave = tid >> 5, lane = tid & 31;
  const int half = (lane >> 4) & 1, nn = lane & 15;
  unsigned* cnt = wsU + 0; unsigned* gen = wsU + 1;
  const int T = seq[0];
  float* probs = out;
  float* o_sel = out + (size_t)T * V;
  float* o_val = o_sel + T;
  float* o_hit = o_val + T;
  float* hist  = o_hit + T;
  const int ntiles = V >> 4;                // V = 100000 -> 6250 tiles of 16 rows

  for (int t = 0; t < T; ++t) {
    float* ptt = probs + (size_t)t * V;

    // ---- Phase A (block 0): GRU(x, h=0) x2, gate, cell, value --------------
    if (blk == 0) {
      float* xls = mv;        // [256]
      float* gx  = mv + 256;  // [768]
      if (tid < HDIM) xls[tid] = wsF[WS_XIN + tid];
      __syncthreads();
      for (int l = 0; l < 2; ++l) {
        for (int r = tid; r < 768; r += NT) {
          const float* wrow = Wih + ((size_t)l * 768 + r) * HDIM;
          float s = bih[l * 768 + r];
          for (int h = 0; h < HDIM; ++h) s = fmaf(wrow[h], xls[h], s);
          gx[r] = s;
        }
        __syncthreads();
        float xnew = 0.f;
        if (tid < HDIM) {
          float rr  = sigmf(gx[tid]       + bhh[l * 768 + tid]);
          float zz  = sigmf(gx[256 + tid] + bhh[l * 768 + 256 + tid]);
          float nnv = tanhf(gx[512 + tid] + rr * bhh[l * 768 + 512 + tid]);
          xnew = (1.f - zz) * nnv;          // h' = (1-z)*n
        }
        __syncthreads();
        if (tid < HDIM) xls[tid] = xnew;
        __syncthreads();
      }
      float cj = 0.f;
      if (tid < HDIM) {
        const float* wrow = fgcW + (size_t)tid * HDIM;
        float s = 0.f;
        for (int h = 0; h < HDIM; ++h) s = fmaf(wrow[h], xls[h], s);
        float g = sigmf(s - wsF[WS_MG + tid]);
        cj = g * wsF[WS_PROC + tid] + (1.f - g) * xls[tid];
        wsF[WS_CELL + tid] = cj;
      }
      rv[tid] = (tid < HDIM) ? cj * valW[tid] : 0.f;
      __syncthreads();
      for (int s = NT/2; s > 0; s >>= 1) { if (tid < s) rv[tid] += rv[tid + s]; __syncthreads(); }
      if (tid == 0) o_val[t] = rv[0] + valb[0];
    }
    grid_sync(cnt, gen);

    // ---- Phase B (all blocks): logits via V_WMMA_F32_16X16X4_F32 -----------
    // A = cell chunk broadcast to all 16 M-rows (every lane's acc[0] == logit[nn]).
    // B = 16 fc_W rows staged coalesced into LDS, read back in B layout.
    if (tid < HDIM) cls[tid] = wsF[WS_CELL + tid];
    __syncthreads();
    float lmax = NEG_INF;
    float* st = stage + wave * (16 * PITCH);
    const int gw = blk * 16 + wave;
    for (int tt = gw; tt < ntiles; tt += NB * 16) {
      const int v0 = tt << 4;
      v8f acc = {0.f, 0.f, 0.f, 0.f, 0.f, 0.f, 0.f, 0.f};
      for (int kc = 0; kc < 8; ++kc) {
        const int k0 = kc << 5;
        const float* gsrc = fcW + (size_t)v0 * HDIM + k0;
        #pragma unroll
        for (int j = 0; j < 4; ++j) {                 // 512 floats, b128 coalesced
          int f4 = lane + j * 32;
          int row = f4 >> 3, c4 = (f4 & 7) << 2;
          v4f d = *(const v4f*)(gsrc + (size_t)row * HDIM + c4);
          *(v4f*)(st + row * PITCH + c4) = d;
        }
        asm volatile("s_wait_dscnt 0" ::: "memory");  // wave-local stage->consume ordering
        #pragma unroll
        for (int kk = 0; kk < 32; kk += 4) {
          v2f a = *(const v2f*)(cls + k0 + kk + half * 2);            // A: K = half*2, half*2+1
          v2f b = *(const v2f*)(st + nn * PITCH + kk + half * 2);     // B: col nn, same K pair
          acc = __builtin_amdgcn_wmma_f32_16x16x4_f32(false, a, false, b,
                                                      (short)0, acc, false, false);
        }
      }
      float logit = acc[0] + fcb[v0 + nn];
      if (lane < 16) ptt[v0 + nn] = logit;
      lmax = fmaxf(lmax, logit);
      int tn = tt + NB * 16;
      if (tn < ntiles) __builtin_prefetch(fcW + (size_t)(tn << 4) * HDIM, 0, 1);
    }
    rv[tid] = lmax;
    __syncthreads();
    for (int s = NT/2; s > 0; s >>= 1) { if (tid < s) rv[tid] = fmaxf(rv[tid], rv[tid + s]); __syncthreads(); }
    if (tid == 0) wsF[WS_BMAX + blk] = rv[0];
    grid_sync(cnt, gen);

    // ---- Phase C (all blocks): exp in-place, partial sum, block top-10 -----
    float gmax = NEG_INF;
    for (int b2 = 0; b2 < NB; ++b2) gmax = fmaxf(gmax, wsF[WS_BMAX + b2]);
    #pragma unroll
    for (int j = 0; j < 4; ++j) { mv[tid + j * NT] = NEG_INF; mi[tid + j * NT] = 0x7fffffff; }
    float esum = 0.f;
    {
      int j = 0;
      for (int i = blk * NT + tid; i < V; i += NB * NT, ++j) {
        float e = __expf(ptt[i] - gmax);
        ptt[i] = e;                                   // unnormalized prob, fixed in Phase D
        esum += e;
        mv[tid + j * NT] = (hist[i] > 0.f) ? e : NEG_INF;  // masked (monotone in prob)
        mi[tid + j * NT] = i;
      }
    }
    rv[tid] = esum;
    __syncthreads();
    for (int s = NT/2; s > 0; s >>= 1) { if (tid < s) rv[tid] += rv[tid + s]; __syncthreads(); }
    if (tid == 0) wsF[WS_BSUM + blk] = rv[0];
    __syncthreads();
    argmax_rounds(mv, mi, rv, ri, rs, 2048, 10,
                  wsF + WS_TKV + blk * 10, ((int*)(wsF + WS_TKI)) + blk * 10);
    grid_sync(cnt, gen);

    // ---- Phase D: merge top-k + select (block 0), normalize probs (all) ----
    float Z = 0.f;
    for (int b2 = 0; b2 < NB; ++b2) Z += wsF[WS_BSUM + b2];
    if (blk == 0) {
      const float* tkv = wsF + WS_TKV;
      const int*   tki = (const int*)(wsF + WS_TKI);
      for (int s = tid; s < 2048; s += NT) {
        if (s < NB * 10) { mv[s] = tkv[s]; mi[s] = tki[s]; }
        else             { mv[s] = NEG_INF; mi[s] = 0x7fffffff; }
      }
      __syncthreads();
      argmax_rounds(mv, mi, rv, ri, rs, 2048, 10, wvs, wis);
      if (tid == 0) {
        // score = hit ? ui*probs (nonzero only at interesting top-k) : masked
        // => argmax is decidable from the 10 candidates alone.
        float hit = 0.f; float bv = NEG_INF; int bi = 0x7fffffff;
        for (int r = 0; r < 10; ++r) {
          int idx = wis[r]; float v = wvs[r];
          if (v <= 0.5f * NEG_INF) continue;
          if (ui[idx] != 0.f) {
            hit = 1.f;
            if (v > bv || (v == bv && idx < bi)) { bv = v; bi = idx; }
          }
        }
        int sel = (hit > 0.f) ? bi : wis[0];
        if (hit > 0.f) hist[sel] = 0.f;               // next_hist = hist - hit*onehot
        o_sel[t] = (float)sel;
        o_hit[t] = hit;
        sI[0] = sel; sF[0] = hit;
      }
      __syncthreads();
      if (tid < HDIM) {
        float h = sF[0]; int sel = sI[0];
        wsF[WS_XIN + tid] = (2.f * h - 1.f) * emb[(size_t)sel * HDIM + tid];
      }
    }
    float invZ = 1.f / Z;
    for (int i = blk * NT + tid; i < V; i += NB * NT) ptt[i] *= invZ;
    grid_sync(cnt, gen);
  }
}

extern "C" void kernel_launch(void* const* d_in, const int* in_sizes, int n_in,
                              void* d_out, int out_size, void* d_ws, size_t ws_size,
                              hipStream_t stream)
{
  const float* ui   = (const float*)d_in[0];
  const float* um   = (const float*)d_in[1];
  const float* mem  = (const float*)d_in[2];
  const float* emb  = (const float*)d_in[3];
  const float* Wih  = (const float*)d_in[4];
  // d_in[5] = gru_Whh: unused (hidden state is always zero)
  const float* bih  = (const float*)d_in[6];
  const float* bhh  = (const float*)d_in[7];
  const float* fcW  = (const float*)d_in[8];
  const float* fcb  = (const float*)d_in[9];
  const float* valW = (const float*)d_in[10];
  const float* valb = (const float*)d_in[11];
  const float* memW = (const float*)d_in[12];
  const float* fgmW = (const float*)d_in[13];
  const float* fgmb = (const float*)d_in[14];
  const float* fgcW = (const float*)d_in[15];
  const int*   seq  = (const int*)d_in[16];
  const int V = in_sizes[0];          // B==1 -> 100000
  float*    wsF = (float*)d_ws;
  unsigned* wsU = (unsigned*)d_ws;
  float*    out = (float*)d_out;

  ppo_init<<<(V + 511) / 512, 512, 0, stream>>>(um, emb, seq, wsF, wsU, out, V);
  ppo_procmem<<<HDIM, 256, 0, stream>>>(mem, memW, wsF, V);
  ppo_mg<<<1, HDIM, 0, stream>>>(fgmW, fgmb, wsF);
  ppo_main<<<NB, NT, 0, stream>>>(ui, emb, Wih, bih, bhh, fcW, fcb, valW, valb,
                                  fgcW, seq, wsF, wsU, out, V);
}